// DetermPolicy_30975304139051
// MI455X (gfx1250) — compile-verified
//
#include <hip/hip_runtime.h>
#include <hip/hip_bf16.h>

#define BDIM   8192
#define OBSDIM 256
#define H1DIM  512
#define UDIM   256
#define SDIM   128
#define MDIM   32

typedef float v2f __attribute__((ext_vector_type(2)));
typedef float v8f __attribute__((ext_vector_type(8)));

__device__ __forceinline__ float softplus_f(float x) {
    // stable log(1+exp(x))
    if (x > 20.0f) return x;
    return log1pf(expf(x));
}

// ---------------------------------------------------------------------------
// Prep: fold softplus/masks/erev into effective-weight tables, gl, cm_t.
// ---------------------------------------------------------------------------
__global__ __launch_bounds__(256) void ltc_prep_kernel(
    const float* __restrict__ w, const float* __restrict__ erev,
    const float* __restrict__ smask,
    const float* __restrict__ sens_w, const float* __restrict__ sens_erev,
    const float* __restrict__ sens_mask,
    const float* __restrict__ gleak, const float* __restrict__ cm,
    float* __restrict__ wp, float* __restrict__ wpe,
    float* __restrict__ swp, float* __restrict__ swpe,
    float* __restrict__ gl, float* __restrict__ cmt)
{
    int i = blockIdx.x * blockDim.x + threadIdx.x;
    if (i < UDIM * SDIM) {
        float v = softplus_f(sens_w[i]) * sens_mask[i];
        swp[i]  = v;
        swpe[i] = v * sens_erev[i];
    }
    if (i < SDIM * SDIM) {
        float v = softplus_f(w[i]) * smask[i];
        wp[i]  = v;
        wpe[i] = v * erev[i];
    }
    if (i < SDIM) {
        gl[i]  = softplus_f(gleak[i]);
        cmt[i] = softplus_f(cm[i]) * 6.0f;   // sp(cm) / (1.0 / 6)
    }
}

// ---------------------------------------------------------------------------
// FP32 WMMA GEMM:  C[M,N] = epilogue(A[M,K] @ B[K,N] + bias)
// Block = 256 threads = 8 waves laid out 2x4; block tile 32(M) x 64(N).
// A tile staged in LDS (pad 4 floats -> bank-conflict-free column reads).
// Uses V_WMMA_F32_16X16X4_F32 (full fp32 precision).
// ---------------------------------------------------------------------------
template <int KDIM, int NDIM, bool RELU, bool INMAP>
__global__ __launch_bounds__(256) void gemm_wmma_f32(
    const float* __restrict__ A, const float* __restrict__ Bm,
    const float* __restrict__ bias,
    const float* __restrict__ iw, const float* __restrict__ ib,
    float* __restrict__ C)
{
    constexpr int LDA = KDIM + 4;          // stride%64 banks == 4 -> conflict free
    extern __shared__ float As[];          // [32][LDA]

    const int tid  = threadIdx.x;
    const int row0 = blockIdx.x * 32;
    const int col0 = blockIdx.y * 64;

    // Cooperative, coalesced load of the 32 x KDIM A tile.
    for (int i = tid; i < 32 * KDIM; i += 256) {
        int r = i / KDIM;
        int c = i - r * KDIM;
        As[r * LDA + c] = A[(row0 + r) * KDIM + c];
    }
    __syncthreads();

    const int lane = tid & 31;
    const int wid  = tid >> 5;
    const int wm   = wid >> 2;             // 0..1  (M wave tile)
    const int wn   = wid & 3;              // 0..3  (N wave tile)
    const int half = lane >> 4;            // 0/1
    const int l16  = lane & 15;
    const int klo  = half * 2;             // K offset within 4-slice per ISA layout

    const float* Arow = &As[(wm * 16 + l16) * LDA];
    const int    gcol = col0 + wn * 16 + l16;

    v8f acc = {};
#pragma unroll 4
    for (int k = 0; k < KDIM; k += 4) {
        // A fragment (16x4 f32): lane half 0 -> K=k..k+1, half 1 -> K=k+2..k+3
        v2f a = *(const v2f*)(Arow + k + klo);
        // B fragment (4x16 f32): row K striped across lanes, same half split
        v2f b;
        b.x = Bm[(k + klo)     * NDIM + gcol];
        b.y = Bm[(k + klo + 1) * NDIM + gcol];
        acc = __builtin_amdgcn_wmma_f32_16x16x4_f32(
            false, a, false, b, (short)0, acc, false, false);
    }

    // Epilogue: bias, optional ReLU, optional affine input map.
    const float bcol = bias[gcol];
    float w_ = 1.0f, o_ = 0.0f;
    if (INMAP) { w_ = iw[gcol]; o_ = ib[gcol]; }
#pragma unroll
    for (int r = 0; r < 8; ++r) {
        float v = acc[r] + bcol;
        if (RELU)  v = fmaxf(v, 0.0f);
        if (INMAP) v = v * w_ + o_;
        // C/D layout: lanes 0-15 -> M=r, lanes 16-31 -> M=r+8
        C[(row0 + wm * 16 + half * 8 + r) * NDIM + gcol] = v;
    }
}

// ---------------------------------------------------------------------------
// Fused LTC cell: sensory reduction + 6 semi-implicit unfolds + tanh output.
// Block = 128 threads (thread = post-neuron s), NB batch rows per block.
// ---------------------------------------------------------------------------
template <int NB>
__global__ __launch_bounds__(128) void ltc_cell_kernel(
    const float* __restrict__ x,      // [B, U]
    const float* __restrict__ ssig,   // [U, S]
    const float* __restrict__ smu,    // [U, S]
    const float* __restrict__ swp,    // [U, S]  sp(sw)*mask
    const float* __restrict__ swpe,   // [U, S]  sp(sw)*mask*erev
    const float* __restrict__ sigma,  // [S, S]
    const float* __restrict__ mu,     // [S, S]
    const float* __restrict__ wp,     // [S, S]  sp(w)*mask
    const float* __restrict__ wpe,    // [S, S]  sp(w)*mask*erev
    const float* __restrict__ gl,     // [S]
    const float* __restrict__ cmt,    // [S]
    const float* __restrict__ vleak,  // [S]
    const float* __restrict__ ow,     // [M]
    const float* __restrict__ ob,     // [M]
    const float* __restrict__ hi,     // [M]
    const float* __restrict__ lo,     // [M]
    float* __restrict__ out)          // [B, M]
{
    __shared__ float xs[NB][UDIM];
    __shared__ float vs[NB][SDIM];

    const int s  = threadIdx.x;          // post-neuron index, 0..127
    const int b0 = blockIdx.x * NB;

    // Stage x rows (coalesced) and init v = 0.
    for (int i = s; i < NB * UDIM; i += SDIM) {
        int b = i >> 8;                  // UDIM == 256
        int u = i & (UDIM - 1);
        xs[b][u] = x[(b0 + b) * UDIM + u];
    }
#pragma unroll
    for (int b = 0; b < NB; ++b) vs[b][s] = 0.0f;
    __syncthreads();

    // Sensory synapses (constant across unfolds).
    float wnum[NB], wden[NB];
#pragma unroll
    for (int b = 0; b < NB; ++b) { wnum[b] = 0.0f; wden[b] = 0.0f; }
    for (int u = 0; u < UDIM; ++u) {
        const int idx = u * SDIM + s;    // coalesced across s
        float sg_ = ssig[idx];
        float mu_ = smu[idx];
        float we_ = swp[idx];
        float wee_ = swpe[idx];
#pragma unroll
        for (int b = 0; b < NB; ++b) {
            float t   = sg_ * (xs[b][u] - mu_);
            float sig = 1.0f / (1.0f + __expf(-t));
            wnum[b] += wee_ * sig;
            wden[b] += we_  * sig;
        }
    }

    const float gl_  = gl[s];
    const float cmt_ = cmt[s];
    const float glv  = gl_ * vleak[s];

    float v[NB];
#pragma unroll
    for (int b = 0; b < NB; ++b) v[b] = 0.0f;

    // 6 semi-implicit ODE unfolds.
    for (int it = 0; it < 6; ++it) {
        float na[NB], da[NB];
#pragma unroll
        for (int b = 0; b < NB; ++b) { na[b] = 0.0f; da[b] = 0.0f; }
        for (int p = 0; p < SDIM; ++p) {
            const int idx = p * SDIM + s;   // coalesced across s
            float sg_  = sigma[idx];
            float mu_  = mu[idx];
            float wp_  = wp[idx];
            float wpe_ = wpe[idx];
#pragma unroll
            for (int b = 0; b < NB; ++b) {
                float t   = sg_ * (vs[b][p] - mu_);   // LDS broadcast read
                float sig = 1.0f / (1.0f + __expf(-t));
                na[b] += wpe_ * sig;
                da[b] += wp_  * sig;
            }
        }
        __syncthreads();                  // all reads of vs done
#pragma unroll
        for (int b = 0; b < NB; ++b) {
            float num = cmt_ * v[b] + glv + na[b] + wnum[b];
            float den = cmt_ + gl_ + da[b] + wden[b] + 1e-8f;
            v[b] = num / den;
            vs[b][s] = v[b];
        }
        __syncthreads();                  // writes visible for next unfold
    }

    // Output epilogue: first M neurons -> affine -> tanh squash.
    if (s < MDIM) {
        float ow_ = ow[s], ob_ = ob[s];
        float h_  = hi[s], l_ = lo[s];
        float a = 0.5f * (h_ - l_), c = 0.5f * (h_ + l_);
#pragma unroll
        for (int b = 0; b < NB; ++b) {
            out[(b0 + b) * MDIM + s] = a * tanhf(v[b] * ow_ + ob_) + c;
        }
    }
}

// ---------------------------------------------------------------------------
extern "C" void kernel_launch(void* const* d_in, const int* in_sizes, int n_in,
                              void* d_out, int out_size, void* d_ws, size_t ws_size,
                              hipStream_t stream) {
    const float* obs        = (const float*)d_in[0];
    const float* W1         = (const float*)d_in[1];
    const float* b1         = (const float*)d_in[2];
    const float* W2         = (const float*)d_in[3];
    const float* b2         = (const float*)d_in[4];
    const float* gleak      = (const float*)d_in[5];
    const float* vleak      = (const float*)d_in[6];
    const float* cm         = (const float*)d_in[7];
    const float* sigma      = (const float*)d_in[8];
    const float* mu         = (const float*)d_in[9];
    const float* w          = (const float*)d_in[10];
    const float* erev       = (const float*)d_in[11];
    const float* ssig       = (const float*)d_in[12];
    const float* smu        = (const float*)d_in[13];
    const float* sens_w     = (const float*)d_in[14];
    const float* sens_erev  = (const float*)d_in[15];
    const float* smask      = (const float*)d_in[16];
    const float* sens_mask  = (const float*)d_in[17];
    const float* input_w    = (const float*)d_in[18];
    const float* input_b    = (const float*)d_in[19];
    const float* output_w   = (const float*)d_in[20];
    const float* output_b   = (const float*)d_in[21];
    const float* act_hi     = (const float*)d_in[22];
    const float* act_lo     = (const float*)d_in[23];
    float* out = (float*)d_out;

    // Workspace layout (floats).
    float* ws   = (float*)d_ws;
    float* h    = ws;                         // 8192*512
    float* x    = h    + BDIM * H1DIM;        // 8192*256
    float* swp  = x    + BDIM * UDIM;         // 256*128
    float* swpe = swp  + UDIM * SDIM;
    float* wp   = swpe + UDIM * SDIM;         // 128*128
    float* wpe  = wp   + SDIM * SDIM;
    float* gl   = wpe  + SDIM * SDIM;         // 128
    float* cmt  = gl   + SDIM;

    // 1) precompute effective weights.
    ltc_prep_kernel<<<128, 256, 0, stream>>>(
        w, erev, smask, sens_w, sens_erev, sens_mask, gleak, cm,
        wp, wpe, swp, swpe, gl, cmt);

    // 2) h = relu(obs @ W1 + b1)   [8192,512]
    {
        dim3 grid(BDIM / 32, H1DIM / 64);
        size_t lds = 32 * (OBSDIM + 4) * sizeof(float);
        gemm_wmma_f32<OBSDIM, H1DIM, true, false><<<grid, 256, lds, stream>>>(
            obs, W1, b1, nullptr, nullptr, h);
    }

    // 3) x = (h @ W2 + b2) * input_w + input_b   [8192,256]
    {
        dim3 grid(BDIM / 32, UDIM / 64);
        size_t lds = 32 * (H1DIM + 4) * sizeof(float);
        gemm_wmma_f32<H1DIM, UDIM, false, true><<<grid, 256, lds, stream>>>(
            h, W2, b2, input_w, input_b, x);
    }

    // 4) fused LTC cell + output squash.
    constexpr int NB = 4;
    ltc_cell_kernel<NB><<<BDIM / NB, SDIM, 0, stream>>>(
        x, ssig, smu, swp, swpe, sigma, mu, wp, wpe,
        gl, cmt, vleak, output_w, output_b, act_hi, act_lo, out);
}